// AttnReadout_65970697667191
// MI455X (gfx1250) — compile-verified
//
#include <hip/hip_runtime.h>
#include <hip/hip_bf16.h>

// ---- problem constants (match reference) ----
static constexpr int N_ = 102400;  // nodes
static constexpr int B_ = 512;     // graphs / segments
static constexpr int D_ = 256;     // feature dim (== output dim)
static constexpr int H_ = 256;     // hidden dim

typedef float v2f __attribute__((ext_vector_type(2)));
typedef float v8f __attribute__((ext_vector_type(8)));

// CDNA5 fp32 WMMA: D(16x16,f32) = A(16x4,f32) x B(4x16,f32) + C
__device__ __forceinline__ v8f wmma_f32_16x16x4(v2f a, v2f b, v8f c) {
    return __builtin_amdgcn_wmma_f32_16x16x4_f32(false, a, false, b,
                                                 (short)0, c, false, false);
}

// gfx1250 async copy global -> LDS (ASYNCcnt-tracked). lds_off is the byte
// offset within the workgroup's LDS allocation; gaddr is a 64-bit global addr.
__device__ __forceinline__ void async_load_b128(unsigned lds_off,
                                                unsigned long long gaddr) {
    asm volatile("global_load_async_to_lds_b128 %0, %1, off"
                 :: "v"(lds_off), "v"(gaddr) : "memory");
}
__device__ __forceinline__ void wait_asynccnt0() {
    asm volatile("s_wait_asynccnt 0x0" ::: "memory");
}

// -------------------------------------------------------------------------
// Kernel P: repack W_u into WMMA B-fragment order so a fragment load is one
// coalesced b64 per lane:
//   frag[(kt*16+n0)*64 + lane*2 + {0,1}] = Wu[(4kt + 2*half + {0,1})*H + n0*16+lc]
// -------------------------------------------------------------------------
__global__ void k_pack(const float* __restrict__ Wu, float* __restrict__ frag) {
    int idx = blockIdx.x * blockDim.x + threadIdx.x;    // [0, 64*16*32)
    if (idx >= 64 * 16 * 32) return;
    int kt   = idx >> 9;
    int rem  = idx & 511;
    int n0   = rem >> 5;
    int lane = rem & 31;
    int half = lane >> 4, lc = lane & 15;
    int row  = 4 * kt + 2 * half;
    int col  = n0 * 16 + lc;
    frag[2 * idx + 0] = Wu[(size_t)row * H_ + col];
    frag[2 * idx + 1] = Wu[(size_t)(row + 1) * H_ + col];
}

// -------------------------------------------------------------------------
// Kernel 0: segment boundaries from sorted seg_ids (lower_bound per segment)
// -------------------------------------------------------------------------
__global__ void k_segstart(const int* __restrict__ seg_ids,
                           int* __restrict__ seg_start) {
    int b = blockIdx.x * blockDim.x + threadIdx.x;
    if (b > B_) return;
    if (b == B_) { seg_start[B_] = N_; return; }
    int lo = 0, hi = N_;
    while (lo < hi) {
        int mid = (lo + hi) >> 1;
        if (seg_ids[mid] < b) lo = mid + 1; else hi = mid;
    }
    seg_start[b] = lo;
}

// -------------------------------------------------------------------------
// Kernel 1: feat_v = feats[last_nodes] @ W_v   [B x H]
// one wave32 per 16x16 output tile, K-loop via V_WMMA_F32_16X16X4_F32
// -------------------------------------------------------------------------
__global__ void k_featv(const float* __restrict__ feats,
                        const long long* __restrict__ last_nodes,
                        const float* __restrict__ Wv,
                        float* __restrict__ featv) {
    const int tile = blockIdx.x;            // (B/16)*(H/16) tiles
    const int rt = tile / (H_ / 16);
    const int ct = tile % (H_ / 16);
    const int lane = threadIdx.x & 31;
    const int half = lane >> 4;             // selects K pair {0,1} vs {2,3}
    const int lc   = lane & 15;

    const long long row = last_nodes[rt * 16 + lc];   // int64 gather
    const float* arow = feats + (size_t)row * D_;
    const int col = ct * 16 + lc;

    v8f acc = {};
    for (int k0 = 0; k0 < D_; k0 += 4) {
        v2f a = *(const v2f*)(arow + k0 + 2 * half);            // A[M=lc][k..k+1]
        v2f b;
        b.x = Wv[(size_t)(k0 + 2 * half)     * H_ + col];       // B[k][n=col]
        b.y = Wv[(size_t)(k0 + 2 * half + 1) * H_ + col];
        acc = wmma_f32_16x16x4(a, b, acc);
    }
#pragma unroll
    for (int j = 0; j < 8; ++j)
        featv[(size_t)(rt * 16 + j + 8 * half) * H_ + col] = acc[j];
}

// -------------------------------------------------------------------------
// Kernel 2 (fused hot path): for each node n
//   u = feats[n] @ W_u ;  e[n] = w_e . sigmoid(u + b_u + feat_v[seg[n]])
// 8 waves/block; wave owns 16 node rows, full 16x256 u-strip in 16 v8f accs.
// W_u fragments stream through 64 KB of LDS in 4 rounds via
// global_load_async_to_lds_b128; all 8 waves consume them with ds_load_b64,
// so each workgroup reads W_u from L2 exactly once.
// -------------------------------------------------------------------------
__global__ void k_e(const float* __restrict__ feats,
                    const int* __restrict__ seg_ids,
                    const float* __restrict__ fragWu,   // packed by k_pack
                    const float* __restrict__ b_u,
                    const float* __restrict__ w_e,
                    const float* __restrict__ featv,
                    float* __restrict__ e_out) {
    __shared__ float ldsW[16384];                      // 64 KB = 16 kt-steps
    const int wave = threadIdx.x >> 5;                 // 0..7
    const int lane = threadIdx.x & 31;
    const int half = lane >> 4;
    const int lc   = lane & 15;
    const int base = blockIdx.x * 128 + wave * 16;     // first node of this wave

    const float* arow = feats + (size_t)(base + lc) * D_;
    const unsigned lds_base = (unsigned)(uintptr_t)(void*)&ldsW[0];

    v8f acc[16];
#pragma unroll
    for (int i = 0; i < 16; ++i) acc[i] = (v8f){};

    for (int q = 0; q < 4; ++q) {                      // 4 staging rounds
        __syncthreads();                               // prev round fully read
        // cooperatively stage 64 KB of the fragment stream (async -> LDS)
#pragma unroll
        for (int i = 0; i < 16; ++i) {
            const int elem = i * 256 + threadIdx.x;    // float4 index [0,4096)
            async_load_b128(lds_base + (unsigned)elem * 16u,
                            (unsigned long long)(uintptr_t)
                                (fragWu + ((size_t)q * 4096 + elem) * 4));
        }
        wait_asynccnt0();
        __syncthreads();                               // LDS tile visible to all

        for (int ktl = 0; ktl < 16; ++ktl) {           // 16 k-steps per round
            const int k0 = (q * 16 + ktl) * 4;
            v2f a = *(const v2f*)(arow + k0 + 2 * half);   // A frag (global)
#pragma unroll
            for (int n0 = 0; n0 < 16; ++n0) {          // 16 independent chains
                v2f b = *(const v2f*)(ldsW + (ktl * 16 + n0) * 64 + lane * 2);
                acc[n0] = wmma_f32_16x16x4(a, b, acc[n0]);
            }
        }
    }

    // epilogue: + b_u + feat_v[seg], sigmoid, dot w_e, reduce 16 lanes -> e
    int segj[8];
#pragma unroll
    for (int j = 0; j < 8; ++j)
        segj[j] = seg_ids[base + j + 8 * half];        // node of C row j

    float part[8];
#pragma unroll
    for (int j = 0; j < 8; ++j) part[j] = 0.0f;

#pragma unroll
    for (int n0 = 0; n0 < 16; ++n0) {
        const int col = n0 * 16 + lc;
        const float bu = b_u[col];
        const float we = w_e[col];
#pragma unroll
        for (int j = 0; j < 8; ++j) {
            float u = acc[n0][j] + bu + featv[(size_t)segj[j] * H_ + col];
            float s = 1.0f / (1.0f + __expf(-u));      // v_exp_f32 (TRANS)
            part[j] += s * we;
        }
    }

#pragma unroll
    for (int j = 0; j < 8; ++j) {
        float v = part[j];
#pragma unroll
        for (int m = 1; m < 16; m <<= 1)
            v += __shfl_xor(v, m, 16);                 // reduce within each half
        if (lc == 0)
            e_out[base + j + 8 * half] = v;
    }
}

// -------------------------------------------------------------------------
// Kernel 3: per-segment max and sum(exp(e - max))
// -------------------------------------------------------------------------
__global__ void k_segstats(const float* __restrict__ e,
                           const int* __restrict__ seg_start,
                           float* __restrict__ seg_max,
                           float* __restrict__ seg_sum) {
    const int b = blockIdx.x;
    const int tid = threadIdx.x;
    const int start = seg_start[b], end = seg_start[b + 1];
    __shared__ float red[256];

    float m = -INFINITY;
    for (int i = start + tid; i < end; i += 256) m = fmaxf(m, e[i]);
    red[tid] = m; __syncthreads();
    for (int s = 128; s > 0; s >>= 1) {
        if (tid < s) red[tid] = fmaxf(red[tid], red[tid + s]);
        __syncthreads();
    }
    const float segm = red[0];
    __syncthreads();

    float su = 0.0f;
    for (int i = start + tid; i < end; i += 256) su += __expf(e[i] - segm);
    red[tid] = su; __syncthreads();
    for (int s = 128; s > 0; s >>= 1) {
        if (tid < s) red[tid] += red[tid + s];
        __syncthreads();
    }
    if (tid == 0) { seg_max[b] = segm; seg_sum[b] = red[0]; }
}

// -------------------------------------------------------------------------
// Kernel 4: rst[b,d] = sum_n alpha[n] * feats[n,d]; weights staged in LDS
// -------------------------------------------------------------------------
__global__ void k_pool(const float* __restrict__ feats,
                       const float* __restrict__ e,
                       const int* __restrict__ seg_start,
                       const float* __restrict__ seg_max,
                       const float* __restrict__ seg_sum,
                       float* __restrict__ out) {
    const int b = blockIdx.x;
    const int tid = threadIdx.x;                       // output column d
    const int start = seg_start[b], end = seg_start[b + 1];
    const float m = seg_max[b];
    const float ssum = seg_sum[b];
    const float inv = (ssum > 0.0f) ? 1.0f / ssum : 0.0f;

    __shared__ float wsh[256];
    float acc = 0.0f;
    for (int chunk = start; chunk < end; chunk += 256) {
        __syncthreads();
        const int n = chunk + tid;
        wsh[tid] = (n < end) ? __expf(e[n] - m) * inv : 0.0f;
        __syncthreads();
        // speculative prefetch of next chunk's rows (global_prefetch_b8)
        __builtin_prefetch(feats + (size_t)(chunk + 256) * D_ + tid, 0, 3);
        const int cnt = min(256, end - chunk);
        for (int i = 0; i < cnt; ++i)
            acc += wsh[i] * feats[(size_t)(chunk + i) * D_ + tid];  // coalesced
    }
    out[(size_t)b * D_ + tid] = acc;                   // [B,1,D] flat; writes all
}

// -------------------------------------------------------------------------
extern "C" void kernel_launch(void* const* d_in, const int* in_sizes, int n_in,
                              void* d_out, int out_size, void* d_ws, size_t ws_size,
                              hipStream_t stream) {
    (void)in_sizes; (void)n_in; (void)out_size; (void)ws_size;
    const float*     feats      = (const float*)d_in[0];      // [N,D] f32
    const long long* last_nodes = (const long long*)d_in[1];  // [B] int64
    const int*       seg_ids    = (const int*)d_in[2];        // [N] int32 (sorted)
    const float*     Wu         = (const float*)d_in[3];      // [D,H]
    const float*     bu         = (const float*)d_in[4];      // [H]
    const float*     Wv         = (const float*)d_in[5];      // [D,H]
    const float*     we         = (const float*)d_in[6];      // [H]
    float* out = (float*)d_out;                               // [B*1*D]

    // workspace layout
    float* featv     = (float*)d_ws;                          // B*H floats
    float* e         = featv + (size_t)B_ * H_;               // N floats
    int*   seg_start = (int*)(e + N_);                        // B+1 ints
    float* seg_max   = (float*)(seg_start + (B_ + 1));        // B floats
    float* seg_sum   = seg_max + B_;                          // B floats
    float* fragWu    = seg_sum + B_;                          // 64*16*64 = 65536 floats

    k_pack<<<(64 * 16 * 32 + 255) / 256, 256, 0, stream>>>(Wu, fragWu);
    k_segstart<<<(B_ + 1 + 255) / 256, 256, 0, stream>>>(seg_ids, seg_start);
    k_featv<<<(B_ / 16) * (H_ / 16), 32, 0, stream>>>(feats, last_nodes, Wv, featv);
    k_e<<<N_ / 128, 256, 0, stream>>>(feats, seg_ids, fragWu, bu, we, featv, e);
    k_segstats<<<B_, 256, 0, stream>>>(e, seg_start, seg_max, seg_sum);
    k_pool<<<B_, 256, 0, stream>>>(feats, e, seg_start, seg_max, seg_sum, out);
}